// XGEA_69441031242523
// MI455X (gfx1250) — compile-verified
//
#include <hip/hip_runtime.h>
#include <math.h>

#define N_NODE 50000
#define N_REL  2000
#define N_TRI  1000000
#define HIDC   128
#define OUTW   768   // 6 * HIDC

typedef __attribute__((ext_vector_type(2))) float v2f;
typedef __attribute__((ext_vector_type(8))) float v8f;

__device__ __forceinline__ float wsum32(float v) {
#pragma unroll
    for (int m = 16; m >= 1; m >>= 1) v += __shfl_xor(v, m, 32);
    return v;
}
__device__ __forceinline__ float wmax32(float v) {
#pragma unroll
    for (int m = 16; m >= 1; m >>= 1) v = fmaxf(v, __shfl_xor(v, m, 32));
    return v;
}

// ---------------------------------------------------------------------------
// 1) per-relation L2 norm of rel_emb rows (wave per relation, float4 lanes)
// ---------------------------------------------------------------------------
__global__ void nr_rel_norm(const float* __restrict__ rel_emb,
                            float* __restrict__ norm) {
    int wave = (blockIdx.x * blockDim.x + threadIdx.x) >> 5;
    int lane = threadIdx.x & 31;
    if (wave >= N_REL) return;
    float4 v = ((const float4*)(rel_emb + wave * HIDC))[lane];
    float ss = v.x * v.x + v.y * v.y + v.z * v.z + v.w * v.w;
    ss = wsum32(ss);
    if (lane == 0) norm[wave] = sqrtf(ss);
}

// ---------------------------------------------------------------------------
// 2) dot table GEMM: [N_REL x 128] @ [128 x 4] via V_WMMA_F32_16X16X4_F32.
//    Columns: 0 = attn_e[0], 1 = attn_e[1], 2 = attn_r[0], 3 = attn_r[1];
//    B is zero-padded to N=16 via a multiply mask (no divergent loads, so
//    the unrolled loads pipeline and EXEC stays all-ones throughout).
// ---------------------------------------------------------------------------
__global__ void nr_dot_wmma(const float* __restrict__ rel_emb,
                            const float* __restrict__ attn_e,
                            const float* __restrict__ attn_r,
                            float* __restrict__ dot4) {
    const int m0   = blockIdx.x * 16;
    const int lane = threadIdx.x & 31;
    const int half = lane >> 4;   // 0: K pair {0,1}, 1: K pair {2,3} (A layout)
    const int l15  = lane & 15;

    const float* acol = rel_emb + (m0 + l15) * HIDC;   // A row for this lane
    // B column pointer: lanes >= 4 get a valid dummy (attn_e) and are masked.
    const float bmask = (l15 < 4) ? 1.f : 0.f;
    const float* bcol = attn_e;
    if      (l15 == 1) bcol = attn_e + HIDC;
    else if (l15 == 2) bcol = attn_r;
    else if (l15 == 3) bcol = attn_r + HIDC;

    v8f c = {0.f, 0.f, 0.f, 0.f, 0.f, 0.f, 0.f, 0.f};
#pragma unroll 4
    for (int k0 = 0; k0 < HIDC; k0 += 4) {
        int kb = k0 + half * 2;
        v2f a; a.x = acol[kb];         a.y = acol[kb + 1];
        v2f b; b.x = bmask * bcol[kb]; b.y = bmask * bcol[kb + 1];
        c = __builtin_amdgcn_wmma_f32_16x16x4_f32(false, a, false, b,
                                                  (short)0, c, false, false);
    }
    if (l15 < 4) {
        int mbase = m0 + half * 8;
#pragma unroll
        for (int v = 0; v < 8; ++v)
            dot4[(mbase + v) * 4 + l15] = c[v];
    }
}

// ---------------------------------------------------------------------------
// 3) CSR row pointers from a sorted segment array
// ---------------------------------------------------------------------------
__global__ void nr_rowptr(const int* __restrict__ seg, int* __restrict__ rowptr,
                          int ntri, int nseg) {
    int t = blockIdx.x * blockDim.x + threadIdx.x;
    if (t >= ntri) return;
    int s  = seg[t];
    int sp = (t == 0) ? -1 : seg[t - 1];
    for (int v = sp + 1; v <= s; ++v) rowptr[v] = t;
    if (t == ntri - 1)
        for (int v = s + 1; v <= nseg; ++v) rowptr[v] = ntri;
}

// ---------------------------------------------------------------------------
// 4) segment mean + tanh  ->  out[:, col_off : col_off+128]
//    (wave per node, lane owns 4 channels, 512B coalesced row gathers)
// ---------------------------------------------------------------------------
__global__ void nr_avg_tanh(const int* __restrict__ rowptr,
                            const int* __restrict__ cols,
                            const float* __restrict__ emb,
                            float* __restrict__ out, int col_off) {
    int node = (blockIdx.x * blockDim.x + threadIdx.x) >> 5;
    int lane = threadIdx.x & 31;
    if (node >= N_NODE) return;
    int start = rowptr[node], end = rowptr[node + 1];
    float4 acc = {0.f, 0.f, 0.f, 0.f};
    for (int t = start; t < end; ++t) {
        if (t + 1 < end)
            __builtin_prefetch(emb + cols[t + 1] * HIDC + (lane << 2), 0, 3);
        float4 v = ((const float4*)(emb + cols[t] * HIDC))[lane];
        acc.x += v.x; acc.y += v.y; acc.z += v.z; acc.w += v.w;
    }
    float inv = 1.f / fmaxf((float)(end - start), 1.f);
    float4 o;
    o.x = tanhf(acc.x * inv); o.y = tanhf(acc.y * inv);
    o.z = tanhf(acc.z * inv); o.w = tanhf(acc.w * inv);
    *(float4*)(out + node * OUTW + col_off + (lane << 2)) = o;
}

// ---------------------------------------------------------------------------
// 5) fused ent+rel attention layer (wave per node, deterministic, no atomics)
//    reads feats from out[:, in_*], writes out[:, out_*] (disjoint columns)
// ---------------------------------------------------------------------------
__global__ void nr_layer(const int* __restrict__ rowptr,
                         const int* __restrict__ dst,
                         const int* __restrict__ rid,
                         const float* __restrict__ rval,
                         const float* __restrict__ norm,
                         const float4* __restrict__ dot4,
                         const float* __restrict__ rel_emb,
                         float* __restrict__ out,
                         int in_e, int out_e, int in_r, int out_r, int l) {
    int node = (blockIdx.x * blockDim.x + threadIdx.x) >> 5;
    int lane = threadIdx.x & 31;
    if (node >= N_NODE) return;
    int start = rowptr[node], end = rowptr[node + 1];

    // ---- pass A: segment softmax stats for both attention instances -------
    float me = -INFINITY, mr = -INFINITY;
    for (int t = start + lane; t < end; t += 32) {
        int r = rid[t]; float rv = rval[t];
        float f = rv / fmaxf(rv * norm[r], 1e-12f);
        float4 dt = dot4[r];
        float ae = f * ((l == 0) ? dt.x : dt.y);
        float ar = f * ((l == 0) ? dt.z : dt.w);
        me = fmaxf(me, ae); mr = fmaxf(mr, ar);
    }
    me = wmax32(me); mr = wmax32(mr);
    float se = 0.f, sr = 0.f;
    for (int t = start + lane; t < end; t += 32) {
        int r = rid[t]; float rv = rval[t];
        float f = rv / fmaxf(rv * norm[r], 1e-12f);
        float4 dt = dot4[r];
        se += __expf(f * ((l == 0) ? dt.x : dt.y) - me);
        sr += __expf(f * ((l == 0) ? dt.z : dt.w) - mr);
    }
    se = wsum32(se); sr = wsum32(sr);
    float ise = (se > 0.f) ? 1.f / se : 0.f;
    float isr = (sr > 0.f) ? 1.f / sr : 0.f;

    // ---- pass B: Householder-reflected, softmax-weighted aggregation ------
    float4 acce = {0.f, 0.f, 0.f, 0.f};
    float4 accr = {0.f, 0.f, 0.f, 0.f};
    const int co = lane << 2;
    for (int t = start; t < end; ++t) {
        int d = dst[t], r = rid[t];
        if (t + 1 < end) {   // gfx1250 global_prefetch_b8 on next rows
            int d2 = dst[t + 1], r2 = rid[t + 1];
            __builtin_prefetch(rel_emb + r2 * HIDC + co, 0, 3);
            __builtin_prefetch(out + d2 * OUTW + in_e + co, 0, 3);
            __builtin_prefetch(out + d2 * OUTW + in_r + co, 0, 3);
        }
        float rv = rval[t];
        float f  = rv / fmaxf(rv * norm[r], 1e-12f);
        float4 e4 = ((const float4*)(rel_emb + r * HIDC))[lane];
        float4 u; u.x = f * e4.x; u.y = f * e4.y; u.z = f * e4.z; u.w = f * e4.w;
        const float* frow = out + d * OUTW;
        float4 he = *(const float4*)(frow + in_e + co);
        float4 hr = *(const float4*)(frow + in_r + co);
        float de = he.x * u.x + he.y * u.y + he.z * u.z + he.w * u.w;
        float dr = hr.x * u.x + hr.y * u.y + hr.z * u.z + hr.w * u.w;
        de = wsum32(de); dr = wsum32(dr);
        float4 dt = dot4[r];
        float we = __expf(f * ((l == 0) ? dt.x : dt.y) - me) * ise;
        float wr = __expf(f * ((l == 0) ? dt.z : dt.w) - mr) * isr;
        float ce = 2.f * we * de, cr = 2.f * wr * dr;
        acce.x += we * he.x - ce * u.x;  acce.y += we * he.y - ce * u.y;
        acce.z += we * he.z - ce * u.z;  acce.w += we * he.w - ce * u.w;
        accr.x += wr * hr.x - cr * u.x;  accr.y += wr * hr.y - cr * u.y;
        accr.z += wr * hr.z - cr * u.z;  accr.w += wr * hr.w - cr * u.w;
    }
    float4 oe, orr;
    oe.x = tanhf(acce.x); oe.y = tanhf(acce.y); oe.z = tanhf(acce.z); oe.w = tanhf(acce.w);
    orr.x = tanhf(accr.x); orr.y = tanhf(accr.y); orr.z = tanhf(accr.z); orr.w = tanhf(accr.w);
    *(float4*)(out + node * OUTW + out_e + co) = oe;
    *(float4*)(out + node * OUTW + out_r + co) = orr;
}

// ---------------------------------------------------------------------------
extern "C" void kernel_launch(void* const* d_in, const int* in_sizes, int n_in,
                              void* d_out, int out_size, void* d_ws, size_t ws_size,
                              hipStream_t stream) {
    (void)in_sizes; (void)n_in; (void)out_size; (void)ws_size;
    const float* ent_emb = (const float*)d_in[0];
    const float* rel_emb = (const float*)d_in[1];
    const float* attn_e  = (const float*)d_in[2];
    const float* attn_r  = (const float*)d_in[3];
    const int*   adj     = (const int*)d_in[4];   // [2, T]
    const int*   r_index = (const int*)d_in[5];   // [2, T] (row0 == arange)
    const float* r_val   = (const float*)d_in[6];
    const int*   ent_adj = (const int*)d_in[7];
    const int*   rel_adj = (const int*)d_in[8];
    float* out = (float*)d_out;

    // workspace carve-out (all 256B aligned)
    char* w = (char*)d_ws;
    float* norm  = (float*)w;  w += 8192;                    // 2000 f32
    float* dot4f = (float*)w;  w += 32768;                   // 2000 x float4
    int*   rp_adj = (int*)w;   w += 200192;                  // 50001 i32
    int*   rp_ent = (int*)w;   w += 200192;
    int*   rp_rel = (int*)w;

    const int* adj_src = adj,      *adj_dst = adj + N_TRI;
    const int* rid     = r_index + N_TRI;
    const int* ea_col  = ent_adj + N_TRI;
    const int* ra_col  = rel_adj + N_TRI;

    // 1) rel norms + 2) WMMA dot table
    nr_rel_norm<<<(N_REL * 32 + 255) / 256, 256, 0, stream>>>(rel_emb, norm);
    nr_dot_wmma<<<N_REL / 16, 32, 0, stream>>>(rel_emb, attn_e, attn_r, dot4f);

    // 3) CSR row pointers (all segment arrays are sorted by construction)
    int tb = (N_TRI + 255) / 256;
    nr_rowptr<<<tb, 256, 0, stream>>>(adj_src, rp_adj, N_TRI, N_NODE);
    nr_rowptr<<<tb, 256, 0, stream>>>(ent_adj, rp_ent, N_TRI, N_NODE);
    nr_rowptr<<<tb, 256, 0, stream>>>(rel_adj, rp_rel, N_TRI, N_NODE);

    // 4) feats_0 = tanh(segment mean); write straight into output columns
    int nb = (N_NODE * 32 + 255) / 256;
    nr_avg_tanh<<<nb, 256, 0, stream>>>(rp_ent, ea_col, ent_emb, out, 0);
    nr_avg_tanh<<<nb, 256, 0, stream>>>(rp_rel, ra_col, rel_emb, out, 384);

    // 5) two fused attention layers (ent instance cols 0/128/256,
    //    rel instance cols 384/512/640)
    nr_layer<<<nb, 256, 0, stream>>>(rp_adj, adj_dst, rid, r_val, norm,
                                     (const float4*)dot4f, rel_emb, out,
                                     0, 128, 384, 512, 0);
    nr_layer<<<nb, 256, 0, stream>>>(rp_adj, adj_dst, rid, r_val, norm,
                                     (const float4*)dot4f, rel_emb, out,
                                     128, 256, 512, 640, 1);
}